// CorrectSemanticGNN_7576322310695
// MI455X (gfx1250) — compile-verified
//
#include <hip/hip_runtime.h>
#include <hip/hip_bf16.h>

typedef float v2f __attribute__((ext_vector_type(2)));
typedef float v8f __attribute__((ext_vector_type(8)));

#define NEG_SLOPE 0.2f
#define LN_EPS 1e-5f
#define BM 64
#define BN 64
#define BKT 16

// ------------------------------------------------------------------
// f32 WMMA GEMM: C[M,N] = A[M,K] @ B[K,N] (+ bias[N]).  Row-major.
// Block = 128 threads = 4 waves; each wave owns a 32x32 output strip
// (four 16x16 v_wmma_f32_16x16x4_f32 accumulators -> 2x operand reuse).
// B tile stored transposed in LDS so each fragment is k-contiguous.
// ------------------------------------------------------------------
__global__ __launch_bounds__(128) void gemm_wmma_f32(
    const float* __restrict__ A, const float* __restrict__ B,
    const float* __restrict__ bias, float* __restrict__ C,
    int M, int N, int K) {
  __shared__ float As[BM][BKT + 1];   // [m][k], +1 pad: conflict-free
  __shared__ float Bt[BN][BKT + 1];   // [n][k] (transposed tile)
  const int bm = blockIdx.y * BM;
  const int bn = blockIdx.x * BN;
  const int tid = threadIdx.x;
  const int wave = tid >> 5;          // 0..3
  const int lane = tid & 31;
  const int wm = (wave & 1) * 32;     // wave row offset inside tile
  const int wn = (wave >> 1) * 32;    // wave col offset inside tile
  const int half = lane >> 4;         // lanes 16-31 hold K=2,3 of each 4-step
  const int l = lane & 15;

  v8f acc00 = {0.f, 0.f, 0.f, 0.f, 0.f, 0.f, 0.f, 0.f};
  v8f acc01 = acc00, acc10 = acc00, acc11 = acc00;

  for (int k0 = 0; k0 < K; k0 += BKT) {
    for (int i = tid; i < BM * BKT; i += 128) {
      int r = i >> 4, c = i & 15;
      int gm = bm + r, gk = k0 + c;
      As[r][c] = (gm < M && gk < K) ? A[(long)gm * K + gk] : 0.0f;
    }
    for (int i = tid; i < BKT * BN; i += 128) {
      int r = i >> 6, c = i & 63;    // r = k, c = n (coalesced global read)
      int gk = k0 + r, gn = bn + c;
      Bt[c][r] = (gk < K && gn < N) ? B[(long)gk * N + gn] : 0.0f;
    }
    // prefetch next K-slab into cache while this slab computes
    const int knext = k0 + BKT;
    if (knext < K) {
      if (tid < BM) {
        int gm = bm + tid;
        if (gm < M) __builtin_prefetch(&A[(long)gm * K + knext], 0, 3);
      } else {
        int t = tid - BM;                       // 0..63
        int gk = knext + (t >> 2);
        int gn = bn + (t & 3) * 16;
        if (gk < K && gn < N) __builtin_prefetch(&B[(long)gk * N + gn], 0, 3);
      }
    }
    __syncthreads();
#pragma unroll
    for (int kk = 0; kk < BKT; kk += 4) {
      // frag layout per ISA 7.12.2: VGPR0 = K{0|2}, VGPR1 = K{1|3}
      v2f a0, a1, b0, b1;
      a0.x = As[wm + l][kk + 2 * half];
      a0.y = As[wm + l][kk + 2 * half + 1];
      a1.x = As[wm + 16 + l][kk + 2 * half];
      a1.y = As[wm + 16 + l][kk + 2 * half + 1];
      b0.x = Bt[wn + l][kk + 2 * half];
      b0.y = Bt[wn + l][kk + 2 * half + 1];
      b1.x = Bt[wn + 16 + l][kk + 2 * half];
      b1.y = Bt[wn + 16 + l][kk + 2 * half + 1];
      acc00 = __builtin_amdgcn_wmma_f32_16x16x4_f32(false, a0, false, b0,
                                                    (short)0, acc00, false, false);
      acc01 = __builtin_amdgcn_wmma_f32_16x16x4_f32(false, a0, false, b1,
                                                    (short)0, acc01, false, false);
      acc10 = __builtin_amdgcn_wmma_f32_16x16x4_f32(false, a1, false, b0,
                                                    (short)0, acc10, false, false);
      acc11 = __builtin_amdgcn_wmma_f32_16x16x4_f32(false, a1, false, b1,
                                                    (short)0, acc11, false, false);
    }
    __syncthreads();
  }
  // D layout: VGPR r -> rows (r | r+8), col = lane&15 within half
  const int gm0 = bm + wm + half * 8;
  const int gn0 = bn + wn + l;
  const int gn1 = bn + wn + 16 + l;
  const float bv0 = (bias != nullptr && gn0 < N) ? bias[gn0] : 0.0f;
  const float bv1 = (bias != nullptr && gn1 < N) ? bias[gn1] : 0.0f;
#pragma unroll
  for (int r = 0; r < 8; ++r) {
    int gma = gm0 + r;
    int gmb = gm0 + 16 + r;
    if (gma < M) {
      if (gn0 < N) C[(long)gma * N + gn0] = acc00[r] + bv0;
      if (gn1 < N) C[(long)gma * N + gn1] = acc01[r] + bv1;
    }
    if (gmb < M) {
      if (gn0 < N) C[(long)gmb * N + gn0] = acc10[r] + bv0;
      if (gn1 < N) C[(long)gmb * N + gn1] = acc11[r] + bv1;
    }
  }
}

// ------------------------------------------------------------------
// Per-head edge-attention scalar: k[h] = sum_c W_edge[h*C+c]*att_edge[h*C+c]
// ------------------------------------------------------------------
__global__ void prep_edge_k(const float* __restrict__ W_edge,
                            const float* __restrict__ att_edge,
                            float* __restrict__ kout, int C) {
  int h = threadIdx.x;
  if (h >= 4) return;
  float s = 0.f;
  for (int c = 0; c < C; ++c) s += W_edge[h * C + c] * att_edge[h * C + c];
  kout[h] = s;
}

// a_src[n,h] = sum_c hh[n,h,c]*att_src[h,c]; same for a_dst
__global__ void node_att(const float* __restrict__ hh,
                         const float* __restrict__ att_s,
                         const float* __restrict__ att_d,
                         float* __restrict__ a_s, float* __restrict__ a_d,
                         int Nn, int C) {
  int i = blockIdx.x * blockDim.x + threadIdx.x;
  if (i >= Nn * 4) return;
  int n = i >> 2, h = i & 3;
  const float* row = hh + (long)n * 4 * C + h * C;
  const float* as = att_s + h * C;
  const float* ad = att_d + h * C;
  float s = 0.f, d = 0.f;
  for (int c = 0; c < C; ++c) {
    float v = row[c];
    s += v * as[c];
    d += v * ad[c];
  }
  a_s[i] = s;
  a_d[i] = d;
}

// order-preserving float<->uint for atomicMax on f32 (handles negatives)
__device__ __forceinline__ unsigned f32_enc(float x) {
  unsigned u = __float_as_uint(x);
  return (u & 0x80000000u) ? ~u : (u | 0x80000000u);
}
__device__ __forceinline__ float f32_dec(unsigned u) {
  return (u & 0x80000000u) ? __uint_as_float(u & 0x7fffffffu)
                           : __uint_as_float(~u);
}

// pass 1: raw alpha = leaky_relu(a_src[src]+a_dst[dst]+ea*k[h]); segment max
__global__ void edge_alpha(const int* __restrict__ src, const int* __restrict__ dst,
                           const float* __restrict__ ea,
                           const float* __restrict__ a_s, const float* __restrict__ a_d,
                           const float* __restrict__ kh,
                           float* __restrict__ alpha, unsigned* __restrict__ amax,
                           int E) {
  int e = blockIdx.x * blockDim.x + threadIdx.x;
  if (e >= E) return;
  int s = src[e], d = dst[e];
  float w = ea[e];
  const float4 as4 = *(const float4*)(a_s + (long)s * 4);
  const float4 ad4 = *(const float4*)(a_d + (long)d * 4);
  const float4 k4 = *(const float4*)kh;
  float v[4];
  v[0] = as4.x + ad4.x + w * k4.x;
  v[1] = as4.y + ad4.y + w * k4.y;
  v[2] = as4.z + ad4.z + w * k4.z;
  v[3] = as4.w + ad4.w + w * k4.w;
#pragma unroll
  for (int h = 0; h < 4; ++h) {
    float a = v[h] > 0.f ? v[h] : NEG_SLOPE * v[h];
    alpha[(long)e * 4 + h] = a;
    atomicMax(&amax[(long)d * 4 + h], f32_enc(a));
  }
}

// pass 2: alpha = exp(alpha - max[dst]); segment sum of denominators
__global__ void edge_exp(const int* __restrict__ dst,
                         float* __restrict__ alpha,
                         const unsigned* __restrict__ amax,
                         float* __restrict__ denom, int E) {
  int e = blockIdx.x * blockDim.x + threadIdx.x;
  if (e >= E) return;
  int d = dst[e];
#pragma unroll
  for (int h = 0; h < 4; ++h) {
    float m = f32_dec(amax[(long)d * 4 + h]);
    float ex = __expf(alpha[(long)e * 4 + h] - m);
    alpha[(long)e * 4 + h] = ex;
    atomicAdd(&denom[(long)d * 4 + h], ex);
  }
}

// pass 3: wave per edge; head-mean weighted gather of hh[src] (L2-resident),
// scatter-add into agg[dst]
__global__ __launch_bounds__(256) void edge_agg(
    const int* __restrict__ src, const int* __restrict__ dst,
    const float* __restrict__ alpha, const float* __restrict__ denom,
    const float* __restrict__ hh, float* __restrict__ agg, int E, int C) {
  int wv = (blockIdx.x * blockDim.x + threadIdx.x) >> 5;
  int lane = threadIdx.x & 31;
  if (wv >= E) return;
  int s = src[wv], d = dst[wv];
  const float4 al = *(const float4*)(alpha + (long)wv * 4);
  const float4 dn = *(const float4*)(denom + (long)d * 4);
  float w0 = al.x / (dn.x + 1e-16f);
  float w1 = al.y / (dn.y + 1e-16f);
  float w2 = al.z / (dn.z + 1e-16f);
  float w3 = al.w / (dn.w + 1e-16f);
  const float* hs = hh + (long)s * 4 * C;
  for (int c = lane; c < C; c += 32) {
    float acc = w0 * hs[c] + w1 * hs[C + c] + w2 * hs[2 * C + c] + w3 * hs[3 * C + c];
    atomicAdd(&agg[(long)d * C + c], 0.25f * acc);
  }
}

// out = LN( relu(agg + gat_bias) + res , g, b ); one wave per node (wave32)
__global__ __launch_bounds__(256) void fuse_relu_res_ln(
    const float* __restrict__ agg, const float* __restrict__ gbias,
    const float* __restrict__ res, const float* __restrict__ g,
    const float* __restrict__ b, float* __restrict__ out, int Nn, int C) {
  int wv = (blockIdx.x * blockDim.x + threadIdx.x) >> 5;
  int lane = threadIdx.x & 31;
  if (wv >= Nn) return;
  const int cnt = C >> 5;  // 4 (C=128) or 2 (C=64)
  float vals[4];
  float sum = 0.f;
  for (int i = 0; i < cnt; ++i) {
    int c = lane + (i << 5);
    float v = agg[(long)wv * C + c] + gbias[c];
    v = fmaxf(v, 0.f) + res[(long)wv * C + c];
    vals[i] = v;
    sum += v;
  }
  for (int off = 16; off > 0; off >>= 1) sum += __shfl_xor(sum, off, 32);
  float mu = sum / (float)C;
  float var = 0.f;
  for (int i = 0; i < cnt; ++i) {
    float dv = vals[i] - mu;
    var += dv * dv;
  }
  for (int off = 16; off > 0; off >>= 1) var += __shfl_xor(var, off, 32);
  var /= (float)C;
  float inv = rsqrtf(var + LN_EPS);
  for (int i = 0; i < cnt; ++i) {
    int c = lane + (i << 5);
    out[(long)wv * C + c] = (vals[i] - mu) * inv * g[c] + b[c];
  }
}

// global mean pool: segment sums + counts
__global__ void pool_sum(const float* __restrict__ h, const int* __restrict__ batch,
                         float* __restrict__ sums, float* __restrict__ cnt, int Nn) {
  int i = blockIdx.x * blockDim.x + threadIdx.x;
  if (i >= Nn * 64) return;
  int n = i >> 6, c = i & 63;
  int b = batch[n];
  atomicAdd(&sums[b * 64 + c], h[i]);
  if (c == 0) atomicAdd(&cnt[b], 1.0f);
}

// classifier head: g = sums/cnt; c1=relu(g@W1+b1); LN; logits=c@W2+b2. One block.
__global__ __launch_bounds__(256) void classifier(
    const float* __restrict__ sums, const float* __restrict__ cnt,
    const float* __restrict__ W1, const float* __restrict__ b1,
    const float* __restrict__ lng, const float* __restrict__ lnb,
    const float* __restrict__ W2, const float* __restrict__ b2,
    float* __restrict__ out) {
  __shared__ float gm[32][64];
  __shared__ float c1[32][64];
  __shared__ float mu[32], inv[32];
  int t = threadIdx.x;
  for (int i = t; i < 32 * 64; i += 256) {
    int gg = i >> 6, c = i & 63;
    gm[gg][c] = sums[i] / fmaxf(cnt[gg], 1.0f);
  }
  __syncthreads();
  for (int i = t; i < 32 * 64; i += 256) {
    int gg = i >> 6, c = i & 63;
    float s = b1[c];
    for (int k = 0; k < 64; ++k) s += gm[gg][k] * W1[k * 64 + c];
    c1[gg][c] = fmaxf(s, 0.f);
  }
  __syncthreads();
  if (t < 32) {
    float s = 0.f;
    for (int c = 0; c < 64; ++c) s += c1[t][c];
    float m = s / 64.f;
    float v = 0.f;
    for (int c = 0; c < 64; ++c) {
      float d = c1[t][c] - m;
      v += d * d;
    }
    v /= 64.f;
    mu[t] = m;
    inv[t] = rsqrtf(v + LN_EPS);
  }
  __syncthreads();
  if (t < 64) {
    int gg = t >> 1, j = t & 1;
    float s = b2[j];
    for (int k = 0; k < 64; ++k) {
      float cn = (c1[gg][k] - mu[gg]) * inv[gg] * lng[k] + lnb[k];
      s += cn * W2[k * 2 + j];
    }
    out[gg * 2 + j] = s;
  }
}

static inline void launch_gemm(const float* A, const float* B, const float* bias,
                               float* C, int M, int N, int K, hipStream_t s) {
  dim3 grid((N + BN - 1) / BN, (M + BM - 1) / BM);
  gemm_wmma_f32<<<grid, 128, 0, s>>>(A, B, bias, C, M, N, K);
}

extern "C" void kernel_launch(void* const* d_in, const int* in_sizes, int n_in,
                              void* d_out, int out_size, void* d_ws, size_t ws_size,
                              hipStream_t stream) {
  (void)n_in; (void)out_size; (void)ws_size;
  const float* x = (const float*)d_in[0];
  const int* edge_index = (const int*)d_in[1];
  const float* edge_attr = (const float*)d_in[2];
  const int* batch = (const int*)d_in[3];
  const float* in_W = (const float*)d_in[4];
  const float* in_b = (const float*)d_in[5];
  const float* gatW[3]  = {(const float*)d_in[6],  (const float*)d_in[12], (const float*)d_in[18]};
  const float* gatAS[3] = {(const float*)d_in[7],  (const float*)d_in[13], (const float*)d_in[19]};
  const float* gatAD[3] = {(const float*)d_in[8],  (const float*)d_in[14], (const float*)d_in[20]};
  const float* gatWE[3] = {(const float*)d_in[9],  (const float*)d_in[15], (const float*)d_in[21]};
  const float* gatAE[3] = {(const float*)d_in[10], (const float*)d_in[16], (const float*)d_in[22]};
  const float* gatB[3]  = {(const float*)d_in[11], (const float*)d_in[17], (const float*)d_in[23]};
  const float* resW[3] = {(const float*)d_in[24], (const float*)d_in[26], (const float*)d_in[28]};
  const float* resB[3] = {(const float*)d_in[25], (const float*)d_in[27], (const float*)d_in[29]};
  const float* lnG[3] = {(const float*)d_in[30], (const float*)d_in[32], (const float*)d_in[34]};
  const float* lnB[3] = {(const float*)d_in[31], (const float*)d_in[33], (const float*)d_in[35]};
  const float* clsW1 = (const float*)d_in[36];
  const float* clsB1 = (const float*)d_in[37];
  const float* clsLG = (const float*)d_in[38];
  const float* clsLB = (const float*)d_in[39];
  const float* clsW2 = (const float*)d_in[40];
  const float* clsB2 = (const float*)d_in[41];

  const int E = in_sizes[2];   // 400000
  const int N = in_sizes[3];   // 50000
  const int* src = edge_index;
  const int* dst = edge_index + E;

  // workspace carve-out (256B aligned slabs)
  char* w = (char*)d_ws;
  auto carve = [&](size_t bytes) -> char* {
    char* p = w;
    w += (bytes + 255) & ~(size_t)255;
    return p;
  };
  float* hA     = (float*)carve((size_t)N * 128 * 4);
  float* hB     = (float*)carve((size_t)N * 128 * 4);
  float* hh     = (float*)carve((size_t)N * 512 * 4);   // [N, 4, C] — fits in 192MB L2
  float* res    = (float*)carve((size_t)N * 128 * 4);
  float* agg    = (float*)carve((size_t)N * 128 * 4);
  float* a_s    = (float*)carve((size_t)N * 4 * 4);
  float* a_d    = (float*)carve((size_t)N * 4 * 4);
  unsigned* amx = (unsigned*)carve((size_t)N * 4 * 4);
  float* den    = (float*)carve((size_t)N * 4 * 4);
  float* alpha  = (float*)carve((size_t)E * 4 * 4);
  float* kbuf   = (float*)carve(256);
  float* sums   = (float*)carve(32 * 64 * 4);
  float* cntb   = (float*)carve(32 * 4);

  // input projection: h = x @ in_W + in_b   (K=135, zero-padded in LDS stage)
  launch_gemm(x, in_W, in_b, hA, N, 128, 135, stream);

  float* hcur = hA;
  float* hnxt = hB;
  for (int L = 0; L < 3; ++L) {
    const int C = (L < 2) ? 128 : 64;
    const int HC = 4 * C;
    launch_gemm(hcur, gatW[L], nullptr, hh, N, HC, 128, stream);        // h_heads
    launch_gemm(hcur, resW[L], resB[L], res, N, C, 128, stream);        // residual proj
    prep_edge_k<<<1, 32, 0, stream>>>(gatWE[L], gatAE[L], kbuf, C);
    node_att<<<(N * 4 + 255) / 256, 256, 0, stream>>>(hh, gatAS[L], gatAD[L],
                                                      a_s, a_d, N, C);
    hipMemsetAsync(amx, 0, (size_t)N * 16, stream);
    hipMemsetAsync(den, 0, (size_t)N * 16, stream);
    hipMemsetAsync(agg, 0, (size_t)N * C * 4, stream);
    edge_alpha<<<(E + 255) / 256, 256, 0, stream>>>(src, dst, edge_attr, a_s, a_d,
                                                    kbuf, alpha, amx, E);
    edge_exp<<<(E + 255) / 256, 256, 0, stream>>>(dst, alpha, amx, den, E);
    edge_agg<<<(E + 7) / 8, 256, 0, stream>>>(src, dst, alpha, den, hh, agg, E, C);
    fuse_relu_res_ln<<<(N + 7) / 8, 256, 0, stream>>>(agg, gatB[L], res, lnG[L],
                                                      lnB[L], hnxt, N, C);
    float* tmp = hcur; hcur = hnxt; hnxt = tmp;
  }

  hipMemsetAsync(sums, 0, 32 * 64 * 4, stream);
  hipMemsetAsync(cntb, 0, 32 * 4, stream);
  pool_sum<<<(N * 64 + 255) / 256, 256, 0, stream>>>(hcur, batch, sums, cntb, N);
  classifier<<<1, 256, 0, stream>>>(sums, cntb, clsW1, clsB1, clsLG, clsLB,
                                    clsW2, clsB2, (float*)d_out);
}